// ArgreementRouting_12128987644061
// MI455X (gfx1250) — compile-verified
//
#include <hip/hip_runtime.h>
#include <math.h>

// ---------------------------------------------------------------------------
// CapsNet agreement routing for MI455X (gfx1250, wave32, WMMA).
//   u = einsum('bni,nio->bno', caps_output, W)   -> f32 WMMA 16x16x4 GEMM
//   4x (softmax / weighted sum / squash) + 3 agreement updates -> wave-per-b
// ---------------------------------------------------------------------------

typedef __attribute__((ext_vector_type(2))) float v2f;
typedef __attribute__((ext_vector_type(8))) float v8f;
typedef __attribute__((ext_vector_type(4))) int   i4;

#define IN_CAPS 32
#define OUT_CAPS 10
#define OUT_DIM 16
#define K_DIM 288
#define N_DIM 160          // OUT_CAPS * OUT_DIM
#define MT 128             // batch rows per workgroup
#define KC 32              // K chunk staged in LDS
#define A_PAD 36           // LDS row stride for A (floats): 16B-aligned rows, no bank conflicts
#define B_PAD 34           // LDS col stride for B^T (floats): 8B-aligned, no bank conflicts

// ---- CDNA5 async global->LDS path (ASYNCcnt-tracked), guarded so a missing
// ---- builtin can never break the build.
#if defined(__has_builtin)
#  if __has_builtin(__builtin_amdgcn_global_load_async_to_lds_b32) && \
      __has_builtin(__builtin_amdgcn_global_load_async_to_lds_b128) && \
      __has_builtin(__builtin_amdgcn_s_wait_asynccnt)
#    define USE_ASYNC_LDS 1
#  endif
#endif
#ifndef USE_ASYNC_LDS
#  define USE_ASYNC_LDS 0
#endif

#if USE_ASYNC_LDS
typedef __attribute__((address_space(1))) int GInt;
typedef __attribute__((address_space(3))) int LInt;
typedef __attribute__((address_space(1))) i4  GInt4;
typedef __attribute__((address_space(3))) i4  LInt4;

__device__ __forceinline__ void async_g2l_b32(const void* g, void* l) {
  __builtin_amdgcn_global_load_async_to_lds_b32(
      (GInt*)(unsigned long long)(size_t)g,
      (LInt*)(unsigned int)(size_t)l, 0, 0);
}
__device__ __forceinline__ void async_g2l_b128(const void* g, void* l) {
  __builtin_amdgcn_global_load_async_to_lds_b128(
      (GInt4*)(unsigned long long)(size_t)g,
      (LInt4*)(unsigned int)(size_t)l, 0, 0);
}
#endif

// ---------------------------------------------------------------------------
// Kernel 1: u[b, n, :] = caps_output[b, n, :] @ W[n, :, :]
// Grid: (ceil(B/128), 32 capsules), 256 threads = 8 waves.
// Each wave: 16 rows x 160 cols = 10 WMMA 16x16 tiles, K-loop of 4.
// ---------------------------------------------------------------------------
__global__ __launch_bounds__(256) void caps_u_gemm(
    const float* __restrict__ A,   // [B, 32, 288]
    const float* __restrict__ W,   // [32, 288, 160]
    float* __restrict__ U,         // [B, 32, 160]
    int Btot)
{
  __shared__ float As[MT * A_PAD];       // 128 x 32 (+pad)  18 KB
  __shared__ float Bs[N_DIM * B_PAD];    // 160 x 32 (B^T, +pad) ~21.8 KB

  const int tid  = threadIdx.x;
  const int wave = tid >> 5;
  const int lane = tid & 31;
  const int n    = blockIdx.y;           // input capsule
  const int m0   = blockIdx.x * MT;      // batch-row base

  const float* Wn = W + (size_t)n * (K_DIM * N_DIM);

  v8f acc[OUT_CAPS];
#pragma unroll
  for (int t = 0; t < OUT_CAPS; ++t)
    acc[t] = (v8f){0.f, 0.f, 0.f, 0.f, 0.f, 0.f, 0.f, 0.f};

  const int half = lane >> 4;            // 0: lanes 0-15, 1: lanes 16-31
  const int l16  = lane & 15;
  const int arow = wave * 16 + l16;      // wave's row inside the 128-row tile

  for (int kk = 0; kk < K_DIM; kk += KC) {
    __syncthreads();
#if USE_ASYNC_LDS
    // A chunk [MT x KC] as B128 async copies (4 floats/op), 4 ops/thread.
#pragma unroll
    for (int i = 0; i < (MT * KC) / (256 * 4); ++i) {
      int q = tid + i * 256;             // b128 index
      int r = q >> 3;                    // 8 x b128 per 32-float row
      int k = (q & 7) << 2;
      int row = m0 + r;
      if (row < Btot)
        async_g2l_b128(&A[((size_t)row * IN_CAPS + n) * K_DIM + kk + k],
                       &As[r * A_PAD + k]);
    }
    // W chunk scattered transposed into LDS: B32 async copies.
#pragma unroll
    for (int idx = tid; idx < KC * N_DIM; idx += 256) {
      int k = idx / N_DIM, c = idx - k * N_DIM;
      async_g2l_b32(&Wn[(size_t)(kk + k) * N_DIM + c], &Bs[c * B_PAD + k]);
    }
    __builtin_amdgcn_s_wait_asynccnt(0);
#else
    // Stage A chunk [MT x KC], coalesced along K.
#pragma unroll
    for (int idx = tid; idx < MT * KC; idx += 256) {
      int r = idx >> 5, k = idx & (KC - 1);
      int row = m0 + r;
      As[r * A_PAD + k] =
          (row < Btot) ? A[((size_t)row * IN_CAPS + n) * K_DIM + kk + k] : 0.0f;
    }
    // Stage W chunk transposed [N_DIM x KC], global reads coalesced along N.
#pragma unroll
    for (int idx = tid; idx < KC * N_DIM; idx += 256) {
      int k = idx / N_DIM, c = idx - k * N_DIM;
      Bs[c * B_PAD + k] = Wn[(size_t)(kk + k) * N_DIM + c];
    }
#endif
    // Prefetch the next weight chunk into cache while we compute.
    if (kk + KC < K_DIM)
      __builtin_prefetch(&Wn[(size_t)(kk + KC) * N_DIM + tid * 20], 0, 1);
    __syncthreads();

#pragma unroll
    for (int k2 = 0; k2 < KC; k2 += 4) {
      // A-frag, ISA 32-bit 16x4 layout: lanes 0-15 hold K0/K1, lanes 16-31 K2/K3.
      const int ak = k2 + (half << 1);   // even -> 8B-aligned v2f loads
      v2f afrag = *(const v2f*)&As[arow * A_PAD + ak];
#pragma unroll
      for (int t = 0; t < OUT_CAPS; ++t) {
        // B-frag (4x16) from transposed LDS: contiguous 8-byte read per lane.
        const int bc = t * 16 + l16;
        v2f bfrag = *(const v2f*)&Bs[bc * B_PAD + ak];
        acc[t] = __builtin_amdgcn_wmma_f32_16x16x4_f32(
            false, afrag, false, bfrag, (short)0, acc[t], false, false);
      }
    }
  }

  // C/D layout: lanes 0-15 -> M=j, lanes 16-31 -> M=8+j; N = lane&15.
  const int mbase = m0 + wave * 16 + (half << 3);
#pragma unroll
  for (int t = 0; t < OUT_CAPS; ++t) {
#pragma unroll
    for (int j = 0; j < 8; ++j) {
      int row = mbase + j;
      if (row < Btot)
        U[((size_t)row * IN_CAPS + n) * N_DIM + t * 16 + l16] = acc[t][j];
    }
  }
}

// ---------------------------------------------------------------------------
// Kernel 2: dynamic routing. One wave per batch element, 8 waves per block.
// b-logits in registers (lane = input capsule n), c/s/v/scale in LDS.
// u reads are L2-resident (84 MB < 192 MB L2).
// ---------------------------------------------------------------------------
__global__ __launch_bounds__(256) void caps_routing(
    const float* __restrict__ U,     // [B, 32, 160]
    const float* __restrict__ Bias,  // [32, 10]
    float* __restrict__ V,           // [B, 10, 16]
    int Btot)
{
  __shared__ float c_lds[8][IN_CAPS][OUT_CAPS];
  __shared__ float sv_lds[8][N_DIM];
  __shared__ float scale_lds[8][OUT_CAPS];

  const int wave = threadIdx.x >> 5;
  const int lane = threadIdx.x & 31;     // lane == input capsule index n
  const int b = blockIdx.x * 8 + wave;
  const bool active = (b < Btot);
  const size_t ub = (size_t)(active ? b : 0) * IN_CAPS * N_DIM;

  float brow[OUT_CAPS];
#pragma unroll
  for (int o = 0; o < OUT_CAPS; ++o)
    brow[o] = Bias[lane * OUT_CAPS + o];

  for (int it = 0; it < 4; ++it) {
    if (it > 0) {
      // b[n,o] += <u[n,o,:], v[o,:]>
#pragma unroll
      for (int o = 0; o < OUT_CAPS; ++o) {
        float a = 0.f;
#pragma unroll
        for (int d = 0; d < OUT_DIM; ++d)
          a += U[ub + (size_t)lane * N_DIM + o * 16 + d] * sv_lds[wave][o * 16 + d];
        brow[o] += a;
      }
    }
    // c[n,:] = softmax over output capsules
    float m = brow[0];
#pragma unroll
    for (int o = 1; o < OUT_CAPS; ++o) m = fmaxf(m, brow[o]);
    float e[OUT_CAPS], sum = 0.f;
#pragma unroll
    for (int o = 0; o < OUT_CAPS; ++o) { e[o] = __expf(brow[o] - m); sum += e[o]; }
    float inv = 1.0f / sum;
#pragma unroll
    for (int o = 0; o < OUT_CAPS; ++o) c_lds[wave][lane][o] = e[o] * inv;
    __syncthreads();

    // s[o,d] = sum_n c[n,o] * u[n,o,d] ; each lane owns 5 (o,d) pairs
#pragma unroll
    for (int p = 0; p < 5; ++p) {
      int od = lane + 32 * p;
      int o  = od >> 4;
      float s = 0.f;
      for (int nn = 0; nn < IN_CAPS; ++nn)
        s += c_lds[wave][nn][o] * U[ub + (size_t)nn * N_DIM + od];
      sv_lds[wave][od] = s;
    }
    __syncthreads();

    // squash scale per output capsule
    if (lane < OUT_CAPS) {
      float sq = 0.f;
#pragma unroll
      for (int d = 0; d < OUT_DIM; ++d) {
        float x = sv_lds[wave][lane * 16 + d];
        sq += x * x;
      }
      scale_lds[wave][lane] = sq / ((1.0f + sq) * sqrtf(sq));
    }
    __syncthreads();

    // v = scale * s (in place)
#pragma unroll
    for (int p = 0; p < 5; ++p) {
      int od = lane + 32 * p;
      sv_lds[wave][od] *= scale_lds[wave][od >> 4];
    }
    __syncthreads();
  }

  if (active) {
#pragma unroll
    for (int p = 0; p < 5; ++p) {
      int od = lane + 32 * p;
      V[(size_t)b * N_DIM + od] = sv_lds[wave][od];
    }
  }
}

// ---------------------------------------------------------------------------
extern "C" void kernel_launch(void* const* d_in, const int* in_sizes, int n_in,
                              void* d_out, int out_size, void* d_ws, size_t ws_size,
                              hipStream_t stream) {
  (void)n_in; (void)out_size; (void)ws_size;
  const float* A    = (const float*)d_in[0];  // caps_output [B,32,288]
  const float* W    = (const float*)d_in[1];  // W           [32,288,160]
  const float* Bias = (const float*)d_in[2];  // b           [32,10]
  float* V = (float*)d_out;                   // v           [B,10,16]
  float* U = (float*)d_ws;                    // scratch u   [B,32,160]

  const int Btot = in_sizes[0] / (IN_CAPS * K_DIM);

  dim3 g1((Btot + MT - 1) / MT, IN_CAPS);
  caps_u_gemm<<<g1, dim3(256), 0, stream>>>(A, W, U, Btot);

  const int g2 = (Btot + 7) / 8;
  caps_routing<<<dim3(g2), dim3(256), 0, stream>>>(U, Bias, V, Btot);
}